// REN_50483045598047
// MI455X (gfx1250) — compile-verified
//
#include <hip/hip_runtime.h>

typedef float v2f __attribute__((ext_vector_type(2)));
typedef float v8f __attribute__((ext_vector_type(8)));

#define N_NODES   50000
#define N_EDGES   800000
#define NE_TOT    850000   // + self loops
#define FEAT      1301
#define HIDDEN    256
#define HEADS     8
#define HEAD_DIM  32
#define B_VAR     4096
#define S_SITES   8
#define NEG_SLOPE 0.2f
#define LN_EPS    1e-5f

// ---------- helpers ----------

// monotonic float<->uint encoding so segment-max can use atomicMax(u32)
__device__ __forceinline__ unsigned fenc(float f) {
  unsigned u = __float_as_uint(f);
  return (u & 0x80000000u) ? ~u : (u | 0x80000000u);
}
__device__ __forceinline__ float fdec(unsigned u) {
  return __uint_as_float((u & 0x80000000u) ? (u & 0x7FFFFFFFu) : ~u);
}

__device__ __forceinline__ float wave_sum(float v) {
  #pragma unroll
  for (int m = 16; m >= 1; m >>= 1) v += __shfl_xor(v, m, 32);
  return v;
}

// ---------- WMMA f32 GEMM: C[M,N] = A[M,K] @ B[K,N] (+bias, +act) ----------
// Block = 256 threads = 8 waves. Block tile 128(M) x 64(N); wave w owns rows
// [w*16, w*16+16) and 4 N-subtiles of 16. K chunked by 32 through LDS.
// B is staged fragment-ready: Bs[kgroup][n][kin] so each lane's {K,K+1} pair
// is one contiguous 8-byte ds_load_b64 (conflict-free across the wave).
// ACT: 0 = none, 1 = relu
template <int ACT>
__global__ __launch_bounds__(256) void wmma_gemm_f32(
    const float* __restrict__ A, const float* __restrict__ B,
    const float* __restrict__ bias, float* __restrict__ C,
    int M, int K, int N) {
  __shared__ float As[128 * 34];   // 128x32, stride 34 (conflict-free b64 reads)
  __shared__ float Bs[8 * 64 * 4]; // [kgroup(8)][n(64)][kin(4)]

  const int t = threadIdx.x;
  const int wave = t >> 5;
  const int lane = t & 31;
  const int m0 = blockIdx.x * 128;
  const int n0 = blockIdx.y * 64;

  v8f acc[4];
  #pragma unroll
  for (int i = 0; i < 4; ++i)
    #pragma unroll
    for (int j = 0; j < 8; ++j) acc[i][j] = 0.0f;

  const int am    = lane & 15;           // M (for A) / N (for B) index in tile
  const int khalf = (lane < 16) ? 0 : 2; // ISA: lanes 0-15 hold K{0,1}, 16-31 K{2,3}

  for (int kc = 0; kc < K; kc += 32) {
    // stage A chunk 128x32 (zero-pad M/K edges)
    #pragma unroll
    for (int i = 0; i < 16; ++i) {
      int idx = t + i * 256;             // 0..4095
      int r = idx >> 5, c = idx & 31;
      int gm = m0 + r, gk = kc + c;
      As[r * 34 + c] = (gm < M && gk < K) ? A[(long)gm * K + gk] : 0.0f;
    }
    // stage B chunk 32x64 in fragment-ready interleave
    #pragma unroll
    for (int i = 0; i < 8; ++i) {
      int idx = t + i * 256;             // 0..2047
      int r = idx >> 6, c = idx & 63;    // r = k within chunk, c = n within tile
      int gk = kc + r;
      Bs[(r >> 2) * 256 + c * 4 + (r & 3)] =
          (gk < K) ? B[(long)gk * N + n0 + c] : 0.0f;
    }
    __syncthreads();

    #pragma unroll
    for (int kg = 0; kg < 8; ++kg) {     // k group of 4
      // A 16x4 fragment: lane<16 -> {K0,K1}; lane>=16 -> {K2,K3} (one b64)
      const v2f a = *(const v2f*)&As[(wave * 16 + am) * 34 + kg * 4 + khalf];
      #pragma unroll
      for (int nt = 0; nt < 4; ++nt) {
        // B 4x16 fragment: contiguous pair in interleaved layout (one b64)
        const v2f b = *(const v2f*)&Bs[kg * 256 + (nt * 16 + am) * 4 + khalf];
        acc[nt] = __builtin_amdgcn_wmma_f32_16x16x4_f32(
            false, a, false, b, (short)0, acc[nt], false, false);
      }
    }
    __syncthreads();
  }

  // C/D layout: VGPR r, lane<16 -> row r, lane>=16 -> row r+8; col = lane&15
  const int roff = (lane < 16) ? 0 : 8;
  #pragma unroll
  for (int nt = 0; nt < 4; ++nt) {
    #pragma unroll
    for (int r = 0; r < 8; ++r) {
      int gm = m0 + wave * 16 + roff + r;
      int gn = n0 + nt * 16 + am;
      if (gm < M) {
        float v = acc[nt][r];
        if (bias) v += bias[gn];
        if (ACT == 1) v = v > 0.0f ? v : 0.0f;
        C[(long)gm * N + gn] = v;
      }
    }
  }
}

// ---------- attention coefficients: a_src/a_dst[n,h] = <xw[n,h,:], att[h,:]> ----------
__global__ void attn_coef_kernel(const float* __restrict__ xw,
                                 const float* __restrict__ att_src,
                                 const float* __restrict__ att_dst,
                                 float* __restrict__ asrc,
                                 float* __restrict__ adst) {
  int i = blockIdx.x * blockDim.x + threadIdx.x;
  if (i >= N_NODES * HEADS) return;
  int n = i >> 3, hh = i & 7;
  const float* row = xw + (long)n * HIDDEN + hh * HEAD_DIM;
  float s1 = 0.0f, s2 = 0.0f;
  #pragma unroll
  for (int c = 0; c < HEAD_DIM; ++c) {
    float x = row[c];
    s1 += x * att_src[hh * HEAD_DIM + c];
    s2 += x * att_dst[hh * HEAD_DIM + c];
  }
  asrc[i] = s1;
  adst[i] = s2;
}

__device__ __forceinline__ void edge_endpoints(const int* __restrict__ ei,
                                               int e, int& s, int& d) {
  if (e < N_EDGES) { s = ei[e]; d = ei[N_EDGES + e]; }
  else             { s = d = e - N_EDGES; }  // self loops
}

// ---------- pass 1: segment max over incoming edges ----------
__global__ void edge_max_kernel(const int* __restrict__ ei,
                                const float* __restrict__ asrc,
                                const float* __restrict__ adst,
                                unsigned* __restrict__ menc) {
  int e = blockIdx.x * blockDim.x + threadIdx.x;
  if (e >= NE_TOT) return;
  int s, d; edge_endpoints(ei, e, s, d);
  #pragma unroll
  for (int hh = 0; hh < HEADS; ++hh) {
    float v = asrc[s * HEADS + hh] + adst[d * HEADS + hh];
    v = v > 0.0f ? v : NEG_SLOPE * v;  // leaky_relu
    atomicMax(&menc[d * HEADS + hh], fenc(v));
  }
}

// ---------- pass 2: segment sum of exp(e - max) ----------
__global__ void edge_sum_kernel(const int* __restrict__ ei,
                                const float* __restrict__ asrc,
                                const float* __restrict__ adst,
                                const unsigned* __restrict__ menc,
                                float* __restrict__ denom) {
  int e = blockIdx.x * blockDim.x + threadIdx.x;
  if (e >= NE_TOT) return;
  int s, d; edge_endpoints(ei, e, s, d);
  #pragma unroll
  for (int hh = 0; hh < HEADS; ++hh) {
    float v = asrc[s * HEADS + hh] + adst[d * HEADS + hh];
    v = v > 0.0f ? v : NEG_SLOPE * v;
    float ex = expf(v - fdec(menc[d * HEADS + hh]));
    atomicAdd(&denom[d * HEADS + hh], ex);
  }
}

// ---------- pass 3: weighted message scatter (wave per edge; lane == channel) --
__global__ __launch_bounds__(256) void edge_scatter_kernel(
    const int* __restrict__ ei, const float* __restrict__ xw,
    const float* __restrict__ asrc, const float* __restrict__ adst,
    const unsigned* __restrict__ menc, const float* __restrict__ denom,
    float* __restrict__ agg) {
  const int wave = threadIdx.x >> 5, lane = threadIdx.x & 31;
  const int e = blockIdx.x * 8 + wave;
  if (e >= NE_TOT) return;
  int s, d; edge_endpoints(ei, e, s, d);
  const float* xrow = xw + (long)s * HIDDEN;
  float* arow = agg + (long)d * HIDDEN;
  __builtin_prefetch(xrow, 0, 1);  // global_prefetch_b8
  #pragma unroll
  for (int hh = 0; hh < HEADS; ++hh) {
    float v = asrc[s * HEADS + hh] + adst[d * HEADS + hh];  // broadcast loads
    v = v > 0.0f ? v : NEG_SLOPE * v;
    float ex = expf(v - fdec(menc[d * HEADS + hh]));
    float alpha = ex / (denom[d * HEADS + hh] + 1e-16f);
    atomicAdd(&arow[hh * HEAD_DIM + lane], xrow[hh * HEAD_DIM + lane] * alpha);
  }
}

// ---------- node epilogue: h = LayerNorm(elu(agg + bias) + h) ----------
__global__ __launch_bounds__(256) void node_epilogue_kernel(
    const float* __restrict__ agg, const float* __restrict__ bias,
    const float* __restrict__ ln_g, const float* __restrict__ ln_b,
    float* __restrict__ h) {
  const int wave = threadIdx.x >> 5, lane = threadIdx.x & 31;
  const int n = blockIdx.x * 8 + wave;
  if (n >= N_NODES) return;
  float r[8];
  float s = 0.0f;
  #pragma unroll
  for (int i = 0; i < 8; ++i) {
    int c = lane + i * 32;
    float v = agg[(long)n * HIDDEN + c] + bias[c];
    v = v > 0.0f ? v : (expf(v) - 1.0f);       // elu
    v += h[(long)n * HIDDEN + c];              // residual
    r[i] = v;
    s += v;
  }
  float mu = wave_sum(s) * (1.0f / HIDDEN);
  float vs = 0.0f;
  #pragma unroll
  for (int i = 0; i < 8; ++i) { float d0 = r[i] - mu; vs += d0 * d0; }
  float inv = rsqrtf(wave_sum(vs) * (1.0f / HIDDEN) + LN_EPS);
  #pragma unroll
  for (int i = 0; i < 8; ++i) {
    int c = lane + i * 32;
    h[(long)n * HIDDEN + c] = (r[i] - mu) * inv * ln_g[c] + ln_b[c];
  }
}

// ---------- pooling head ----------
__global__ __launch_bounds__(256) void gather_sites_kernel(
    const int* __restrict__ sites, const float* __restrict__ h,
    float* __restrict__ semb) {
  const int wave = threadIdx.x >> 5, lane = threadIdx.x & 31;
  const int i = blockIdx.x * 8 + wave;
  if (i >= B_VAR * S_SITES) return;
  int node = sites[i];
  #pragma unroll
  for (int j = 0; j < 8; ++j)
    semb[(long)i * HIDDEN + lane + j * 32] = h[(long)node * HIDDEN + lane + j * 32];
}

__global__ __launch_bounds__(256) void score_kernel(
    const float* __restrict__ keys, const float* __restrict__ q,
    float* __restrict__ scores) {
  const int wave = threadIdx.x >> 5, lane = threadIdx.x & 31;
  const int i = blockIdx.x * 8 + wave;
  if (i >= B_VAR * S_SITES) return;
  float s = 0.0f;
  #pragma unroll
  for (int j = 0; j < 8; ++j) { int c = lane + j * 32; s += keys[(long)i * HIDDEN + c] * q[c]; }
  s = wave_sum(s);
  if (lane == 0) scores[i] = s * (1.0f / 16.0f);  // / sqrt(256)
}

__global__ __launch_bounds__(256) void pool_kernel(
    const float* __restrict__ scores, const float* __restrict__ semb,
    float* __restrict__ pooled) {
  const int wave = threadIdx.x >> 5, lane = threadIdx.x & 31;
  const int b = blockIdx.x * 8 + wave;
  if (b >= B_VAR) return;
  float sc[S_SITES], mx = -3.4e38f;
  #pragma unroll
  for (int s = 0; s < S_SITES; ++s) { sc[s] = scores[b * S_SITES + s]; mx = fmaxf(mx, sc[s]); }
  float sum = 0.0f;
  #pragma unroll
  for (int s = 0; s < S_SITES; ++s) { sc[s] = expf(sc[s] - mx); sum += sc[s]; }
  float rs = 1.0f / sum;
  #pragma unroll
  for (int j = 0; j < 8; ++j) {
    int c = lane + j * 32;
    float acc = 0.0f;
    #pragma unroll
    for (int s = 0; s < S_SITES; ++s)
      acc += sc[s] * rs * semb[(long)(b * S_SITES + s) * HIDDEN + c];
    pooled[(long)b * HIDDEN + c] = acc;
  }
}

__global__ __launch_bounds__(256) void mlp_out_kernel(
    const float* __restrict__ hmid, const float* __restrict__ W2,
    const float* __restrict__ b2, float* __restrict__ out) {
  const int wave = threadIdx.x >> 5, lane = threadIdx.x & 31;
  const int b = blockIdx.x * 8 + wave;
  if (b >= B_VAR) return;
  float s = 0.0f;
  #pragma unroll
  for (int j = 0; j < 4; ++j) { int c = lane + j * 32; s += hmid[(long)b * 128 + c] * W2[c]; }
  s = wave_sum(s);
  if (lane == 0) out[b] = s + b2[0];
}

// ---------- host orchestration ----------
extern "C" void kernel_launch(void* const* d_in, const int* in_sizes, int n_in,
                              void* d_out, int out_size, void* d_ws, size_t ws_size,
                              hipStream_t stream) {
  (void)in_sizes; (void)n_in; (void)out_size; (void)ws_size;
  // d_in order = setup_inputs() dict order (params flattened in insertion order):
  // 0:x 1:edge_index 2:sites 3:in_W 4:in_b
  // per layer l (l=0..2): 5+6l:W 6+6l:att_src 7+6l:att_dst 8+6l:bias 9+6l:ln_g 10+6l:ln_b
  // 23:pool_query 24:pool_kW 25:pool_kb 26:mlp_W1 27:mlp_b1 28:mlp_W2 29:mlp_b2
  const float* x       = (const float*)d_in[0];
  const int*   ei      = (const int*)d_in[1];
  const int*   sites   = (const int*)d_in[2];
  const float* in_W    = (const float*)d_in[3];
  const float* in_b    = (const float*)d_in[4];
  const float* pool_q  = (const float*)d_in[23];
  const float* pool_kW = (const float*)d_in[24];
  const float* pool_kb = (const float*)d_in[25];
  const float* mlp_W1  = (const float*)d_in[26];
  const float* mlp_b1  = (const float*)d_in[27];
  const float* mlp_W2  = (const float*)d_in[28];
  const float* mlp_b2  = (const float*)d_in[29];

  char* ws = (char*)d_ws;
  size_t off = 0;
  auto carve = [&](size_t bytes) -> void* {
    void* p = ws + off;
    off += (bytes + 255) & ~(size_t)255;
    return p;
  };
  float*    h      = (float*)carve((size_t)N_NODES * HIDDEN * 4);
  float*    xw     = (float*)carve((size_t)N_NODES * HIDDEN * 4);
  float*    agg    = (float*)carve((size_t)N_NODES * HIDDEN * 4);
  float*    asrc   = (float*)carve((size_t)N_NODES * HEADS * 4);
  float*    adst   = (float*)carve((size_t)N_NODES * HEADS * 4);
  unsigned* menc   = (unsigned*)carve((size_t)N_NODES * HEADS * 4);
  float*    denom  = (float*)carve((size_t)N_NODES * HEADS * 4);
  float*    semb   = (float*)carve((size_t)B_VAR * S_SITES * HIDDEN * 4);
  float*    keys   = (float*)carve((size_t)B_VAR * S_SITES * HIDDEN * 4);
  float*    scores = (float*)carve((size_t)B_VAR * S_SITES * 4);
  float*    pooled = (float*)carve((size_t)B_VAR * HIDDEN * 4);
  float*    hmid   = (float*)carve((size_t)B_VAR * 128 * 4);

  const dim3 blk(256);

  // h = x @ in_W + in_b    [50000,1301]x[1301,256]
  wmma_gemm_f32<0><<<dim3((N_NODES + 127) / 128, HIDDEN / 64), blk, 0, stream>>>(
      x, in_W, in_b, h, N_NODES, FEAT, HIDDEN);

  for (int l = 0; l < 3; ++l) {
    const float* W       = (const float*)d_in[5 + 6 * l];
    const float* att_src = (const float*)d_in[6 + 6 * l];
    const float* att_dst = (const float*)d_in[7 + 6 * l];
    const float* bias    = (const float*)d_in[8 + 6 * l];
    const float* ln_g    = (const float*)d_in[9 + 6 * l];
    const float* ln_b    = (const float*)d_in[10 + 6 * l];

    // xw = h @ W
    wmma_gemm_f32<0><<<dim3((N_NODES + 127) / 128, HIDDEN / 64), blk, 0, stream>>>(
        h, W, nullptr, xw, N_NODES, HIDDEN, HIDDEN);

    hipMemsetAsync(menc,  0, (size_t)N_NODES * HEADS * 4, stream);
    hipMemsetAsync(denom, 0, (size_t)N_NODES * HEADS * 4, stream);
    hipMemsetAsync(agg,   0, (size_t)N_NODES * HIDDEN * 4, stream);

    attn_coef_kernel<<<(N_NODES * HEADS + 255) / 256, blk, 0, stream>>>(
        xw, att_src, att_dst, asrc, adst);
    edge_max_kernel<<<(NE_TOT + 255) / 256, blk, 0, stream>>>(ei, asrc, adst, menc);
    edge_sum_kernel<<<(NE_TOT + 255) / 256, blk, 0, stream>>>(ei, asrc, adst, menc, denom);
    edge_scatter_kernel<<<(NE_TOT + 7) / 8, blk, 0, stream>>>(
        ei, xw, asrc, adst, menc, denom, agg);
    node_epilogue_kernel<<<(N_NODES + 7) / 8, blk, 0, stream>>>(agg, bias, ln_g, ln_b, h);
  }

  // pooling head
  gather_sites_kernel<<<(B_VAR * S_SITES + 7) / 8, blk, 0, stream>>>(sites, h, semb);
  wmma_gemm_f32<0><<<dim3((B_VAR * S_SITES) / 128, HIDDEN / 64), blk, 0, stream>>>(
      semb, pool_kW, pool_kb, keys, B_VAR * S_SITES, HIDDEN, HIDDEN);
  score_kernel<<<(B_VAR * S_SITES + 7) / 8, blk, 0, stream>>>(keys, pool_q, scores);
  pool_kernel<<<(B_VAR + 7) / 8, blk, 0, stream>>>(scores, semb, pooled);
  wmma_gemm_f32<1><<<dim3(B_VAR / 128, 128 / 64), blk, 0, stream>>>(
      pooled, mlp_W1, mlp_b1, hmid, B_VAR, HIDDEN, 128);
  mlp_out_kernel<<<(B_VAR + 7) / 8, blk, 0, stream>>>(hmid, mlp_W2, mlp_b2, (float*)d_out);
}